// JointModel_29506425323522
// MI455X (gfx1250) — compile-verified
//
#include <hip/hip_runtime.h>
#include <hip/hip_bf16.h>

typedef __attribute__((ext_vector_type(2))) float v2f;
typedef __attribute__((ext_vector_type(8))) float v8f;

#define IMGS   128
#define CH     3
#define HW     224
#define PLANE  (HW * HW)          // 50176
#define BH     8                  // output rows per workgroup
#define LROWS  (BH + 2)           // staged rows incl. halo
#define LCOLS  (HW + 2)           // staged cols incl. halo (226)
#define LDS_IN_N   (CH * LROWS * LCOLS)   // 6780
#define LDS_IN_PAD 7024                   // zeroed tail absorbs K=27 dummy reads
#define LDS_OUT_N  (CH * BH * HW)         // 5376

// ws layout (bytes)
#define WS_FEAT_OFF  0                    // 384 floats
#define WS_B_OFF     2048                 // 128 * 448 floats (28x16 B-matrix per image)
#define WS_BIAS_OFF  (2048 + 128*448*4)   // 128 * 16 floats

__device__ __forceinline__ float std_c(int c)  { return c == 0 ? 0.229f : (c == 1 ? 0.224f : 0.225f); }
__device__ __forceinline__ float mean_c(int c) { return c == 0 ? 0.485f : (c == 1 ? 0.456f : 0.406f); }

// k -> LDS offset of (ic, dy, dx); all call sites pass compile-time constants,
// so this folds to an inline literal (no runtime div/mod).
__device__ __forceinline__ constexpr int koff(int kk) {
    return (kk / 9) * (LROWS * LCOLS) + ((kk % 9) / 3) * LCOLS + ((kk % 9) % 3);
}

// ---------------- Kernel 1: feat[n,c] = STD[c]*mean(x[n,c]) + MEAN[c] ----------------
__global__ __launch_bounds__(256) void k_feat(const float* __restrict__ x,
                                              float* __restrict__ feat) {
    __shared__ float red[256];
    const int bid = blockIdx.x;          // 0..383  (n*3 + c)
    const int c   = bid % CH;
    const int tid = threadIdx.x;
    const float* p = x + (size_t)bid * PLANE;
    float s = 0.f;
    for (int j = tid; j < PLANE; j += 256) s += p[j];
    red[tid] = s;
    __syncthreads();
    for (int off = 128; off > 0; off >>= 1) {
        if (tid < off) red[tid] += red[tid + off];
        __syncthreads();
    }
    if (tid == 0)
        feat[bid] = std_c(c) * (red[0] * (1.0f / PLANE)) + mean_c(c);
}

// ------- Kernel 2: per-image kernel -> WMMA B-matrix (28x16, zero padded) + bias16 -------
__global__ __launch_bounds__(128) void k_build(const float* __restrict__ feat,
                                               const float* __restrict__ W1,
                                               const float* __restrict__ b1,
                                               float* __restrict__ Bmat,
                                               float* __restrict__ biasm) {
    const int n   = blockIdx.x;
    const int tid = threadIdx.x;
    for (int e = tid; e < 448 + 16; e += 128) {
        if (e < 448) Bmat[n * 448 + e] = 0.f;
        else         biasm[n * 16 + (e - 448)] = 0.f;
    }
    __syncthreads();
    if (tid < 84) {
        const float f0 = feat[n * 3 + 0], f1 = feat[n * 3 + 1], f2 = feat[n * 3 + 2];
        const float kj = f0 * W1[tid] + f1 * W1[84 + tid] + f2 * W1[168 + tid] + b1[tid];
        if (tid < 81) {
            const int oc = tid / 27;            // j = oc*27 + k
            const int k  = tid % 27;
            Bmat[n * 448 + k * 16 + oc] = kj;   // B[k][oc]; row 27 & oc 3..15 stay 0
        } else {
            biasm[n * 16 + (tid - 81)] = kj;
        }
    }
}

// ---------------- Kernel 3: dynamic 3x3 conv via chained V_WMMA_F32_16X16X4_F32 ----------------
__global__ __launch_bounds__(256) void k_conv(const float* __restrict__ x,
                                              const float* __restrict__ Bmat,
                                              const float* __restrict__ biasm,
                                              float* __restrict__ out) {
    __shared__ float lds_in[LDS_IN_PAD];
    __shared__ float lds_out[LDS_OUT_N];

    const int bid = blockIdx.x;          // 128 * 28
    const int n   = bid / (HW / BH);
    const int h0  = (bid % (HW / BH)) * BH;
    const int tid = threadIdx.x;

    // ---- stage origin rows h0-1 .. h0+BH (halo, affine transform, zero padding) ----
    const float* xn = x + (size_t)n * CH * PLANE;
    for (int e = tid; e < LDS_IN_PAD; e += 256) {
        float v = 0.f;
        if (e < LDS_IN_N) {
            const int ic  = e / (LROWS * LCOLS);
            const int rem = e % (LROWS * LCOLS);
            const int r   = rem / LCOLS;
            const int cc  = rem % LCOLS;
            const int h   = h0 - 1 + r;
            const int w   = cc - 1;
            if (h >= 0 && h < HW && w >= 0 && w < HW)
                v = xn[ic * PLANE + h * HW + w] * std_c(ic) + mean_c(ic);
        }
        lds_in[e] = v;
    }

    // ---- per-lane WMMA operand setup ----
    const int lane = tid & 31;
    const int wid  = tid >> 5;           // wave id = output row within block
    const int half = lane >> 4;
    const int col  = lane & 15;          // pixel (A/M) or oc (B/N) index

    // B operand: lanes 0-15 hold K rows 4s,4s+1 ; lanes 16-31 hold 4s+2,4s+3
    const float* Bn = Bmat + n * 448;
    v2f bm[7];
#pragma unroll
    for (int s = 0; s < 7; ++s) {
        const int k0 = 4 * s + 2 * half;
        bm[s].x = Bn[k0 * 16 + col];
        bm[s].y = Bn[(k0 + 1) * 16 + col];
    }

    // bias -> C init (all 8 rows of this lane share N=col)
    const float cb = biasm[n * 16 + col];
    const v8f cinit = {cb, cb, cb, cb, cb, cb, cb, cb};

    // A gather offsets: select between two compile-time koff() constants on `half`
    // (single v_cndmask each) + rowbase. K=27 (s==6,q==1,half==1) -> zeroed pad.
    const int rowbase = wid * LCOLS + col;   // (hh-1)*LCOLS + pix,  hh = wid+1
    int oX[7], oY[7];
#pragma unroll
    for (int s = 0; s < 7; ++s)
        oX[s] = (half ? koff(4 * s + 2) : koff(4 * s)) + rowbase;
#pragma unroll
    for (int s = 0; s < 6; ++s)
        oY[s] = (half ? koff(4 * s + 3) : koff(4 * s + 1)) + rowbase;
    oY[6] = half ? LDS_IN_N : (koff(25) + rowbase);

    __syncthreads();

    // ---- 14 tiles of 16 pixels per row; 7 chained f32 WMMAs per tile ----
#pragma unroll
    for (int t = 0; t < HW / 16; ++t) {
        const int w0 = t * 16;
        v8f acc = cinit;
#pragma unroll
        for (int s = 0; s < 7; ++s) {
            v2f a;
            a.x = lds_in[oX[s] + w0];
            a.y = lds_in[oY[s] + w0];
            acc = __builtin_amdgcn_wmma_f32_16x16x4_f32(
                false, a, false, bm[s], (short)0, acc, false, false);
        }
        // D: lane<16 holds (M=r, N=col) r=0..7 ; lane>=16 holds (M=8+r, N=col)
        if (col < CH) {
            float* dst = &lds_out[col * (BH * HW) + wid * HW + w0 + half * 8];
#pragma unroll
            for (int r = 0; r < 8; ++r) dst[r] = acc[r];
        }
    }

    __syncthreads();

    // ---- coalesced store of the 3 x 8 x 224 output block ----
    float* on = out + (size_t)n * CH * PLANE;
    for (int e = tid; e < LDS_OUT_N; e += 256) {
        const int oc  = e / (BH * HW);
        const int rem = e % (BH * HW);
        const int r   = rem / HW;
        const int w   = rem % HW;
        on[oc * PLANE + (h0 + r) * HW + w] = lds_out[e];
    }
}

extern "C" void kernel_launch(void* const* d_in, const int* in_sizes, int n_in,
                              void* d_out, int out_size, void* d_ws, size_t ws_size,
                              hipStream_t stream) {
    const float* x  = (const float*)d_in[0];   // (128,3,224,224)
    const float* W1 = (const float*)d_in[1];   // (3,84)
    const float* b1 = (const float*)d_in[2];   // (84,)
    float* out = (float*)d_out;

    char* ws = (char*)d_ws;
    float* feat  = (float*)(ws + WS_FEAT_OFF);
    float* Bmat  = (float*)(ws + WS_B_OFF);
    float* biasm = (float*)(ws + WS_BIAS_OFF);

    k_feat <<<dim3(IMGS * CH), dim3(256), 0, stream>>>(x, feat);
    k_build<<<dim3(IMGS),      dim3(128), 0, stream>>>(feat, W1, b1, Bmat, biasm);
    k_conv <<<dim3(IMGS * (HW / BH)), dim3(256), 0, stream>>>(x, Bmat, biasm, out);
}